// FactorizedTemplate_13168369729892
// MI455X (gfx1250) — compile-verified
//
#include <hip/hip_runtime.h>
#include <hip/hip_bf16.h>
#include <math.h>

#define L 768
#define C 64
#define CT3 192
#define NOUT 1024   // RANK * C_OUT
#define LN_EPS 1e-5f

typedef __attribute__((ext_vector_type(2))) float v2f;
typedef __attribute__((ext_vector_type(8))) float v8f;

// ---------------- Kernel 1: logits[i,j] = feat[i,j,:].Wq + bq ----------------
__global__ void k_logits(const float* __restrict__ feat, const float* __restrict__ Wq,
                         const float* __restrict__ bq, float* __restrict__ logits) {
    int idx = blockIdx.x * blockDim.x + threadIdx.x;      // 0 .. L*L-1
    const float4* f = (const float4*)(feat + (size_t)idx * C);
    float acc = 0.f;
#pragma unroll
    for (int k = 0; k < C / 4; ++k) {
        float4 v = f[k];
        acc += v.x * Wq[4 * k + 0] + v.y * Wq[4 * k + 1] +
               v.z * Wq[4 * k + 2] + v.w * Wq[4 * k + 3];
    }
    logits[idx] = acc + bq[0];
}

// ------------- Kernel 2: softmax stats (max, expsum) per row / col -----------
__global__ void k_stats(const float* __restrict__ logits,
                        float* __restrict__ rmax, float* __restrict__ rsum,
                        float* __restrict__ cmax, float* __restrict__ csum) {
    int line = blockIdx.x;            // row i (y==0) or col j (y==1)
    int colmode = blockIdx.y;
    __shared__ float red[256];
    int t = threadIdx.x;

    float m = -3.4e38f;
    for (int k = t; k < L; k += 256) {
        float v = colmode ? logits[(size_t)k * L + line] : logits[(size_t)line * L + k];
        m = fmaxf(m, v);
    }
    red[t] = m; __syncthreads();
    for (int s = 128; s > 0; s >>= 1) {
        if (t < s) red[t] = fmaxf(red[t], red[t + s]);
        __syncthreads();
    }
    m = red[0]; __syncthreads();

    float sum = 0.f;
    for (int k = t; k < L; k += 256) {
        float v = colmode ? logits[(size_t)k * L + line] : logits[(size_t)line * L + k];
        sum += expf(v - m);
    }
    red[t] = sum; __syncthreads();
    for (int s = 128; s > 0; s >>= 1) {
        if (t < s) red[t] += red[t + s];
        __syncthreads();
    }
    if (t == 0) {
        if (colmode) { cmax[line] = m; csum[line] = red[0]; }
        else         { rmax[line] = m; rsum[line] = red[0]; }
    }
}

// ------- Kernel 3: weighted feature sum (row_feat or col_feat per flag) ------
// 256 threads = 16 channel-quads (float4) x 16-way j/i parallel; b128 loads.
__global__ void k_wsum(const float* __restrict__ feat, const float* __restrict__ logits,
                       const float* __restrict__ mx, const float* __restrict__ sm,
                       float* __restrict__ outfeat, int colmode) {
    int line = blockIdx.x;            // i for row mode, j for col mode
    int t = threadIdx.x;              // 256 threads
    int c4  = t & 15;                 // channels [c4*4, c4*4+3]
    int sub = t >> 4;                 // 0..15
    float m = mx[line];
    float inv_s = 1.0f / sm[line];
    float ax = 0.f, ay = 0.f, az = 0.f, aw = 0.f;
    for (int k = sub; k < L; k += 16) {
        size_t fidx = colmode ? ((size_t)k * L + line) : ((size_t)line * L + k);
        float w = expf(logits[fidx] - m);
        float4 v = *(const float4*)(feat + fidx * C + c4 * 4);
        ax += w * v.x; ay += w * v.y; az += w * v.z; aw += w * v.w;
    }
    __shared__ float4 red[256];
    red[t] = make_float4(ax, ay, az, aw);
    __syncthreads();
    // reduce across sub (stride multiples of 16 preserve c4)
    for (int s = 128; s >= 16; s >>= 1) {
        if (t < s) {
            float4 a = red[t], b = red[t + s];
            red[t] = make_float4(a.x + b.x, a.y + b.y, a.z + b.z, a.w + b.w);
        }
        __syncthreads();
    }
    if (t < 16) {
        float4 v = red[t];
        v.x *= inv_s; v.y *= inv_s; v.z *= inv_s; v.w *= inv_s;
        *(float4*)(outfeat + (size_t)line * C + t * 4) = v;
    }
}

// --------- Kernel 4: diag gather + concat + shared-stats LayerNorm -----------
__global__ void k_ln(const float* __restrict__ feat,
                     const float* __restrict__ rowf, const float* __restrict__ colf,
                     const float* __restrict__ g, const float* __restrict__ b,
                     float* __restrict__ Lm, float* __restrict__ Rm) {
    int i = blockIdx.x;
    int t = threadIdx.x;              // 0..191
    float lv, rv;
    if (t < 64) {
        float d = feat[((size_t)i * L + i) * C + t];
        lv = d; rv = d;
    } else if (t < 128) {
        lv = rowf[(size_t)i * C + (t - 64)];
        rv = colf[(size_t)i * C + (t - 64)];
    } else {
        lv = colf[(size_t)i * C + (t - 128)];
        rv = rowf[(size_t)i * C + (t - 128)];
    }
    __shared__ float red[192];
    // mean (left and right hold the same multiset of values -> shared stats)
    red[t] = lv; __syncthreads();
    if (t < 64) red[t] += red[t + 64] + red[t + 128];
    __syncthreads();
    for (int s = 32; s > 0; s >>= 1) { if (t < s) red[t] += red[t + s]; __syncthreads(); }
    float mu = red[0] * (1.0f / 192.0f); __syncthreads();
    // variance
    float d = lv - mu;
    red[t] = d * d; __syncthreads();
    if (t < 64) red[t] += red[t + 64] + red[t + 128];
    __syncthreads();
    for (int s = 32; s > 0; s >>= 1) { if (t < s) red[t] += red[t + s]; __syncthreads(); }
    float inv = rsqrtf(red[0] * (1.0f / 192.0f) + LN_EPS);
    float gg = g[t], bb = b[t];
    Lm[(size_t)i * CT3 + t] = (lv - mu) * inv * gg + bb;
    Rm[(size_t)i * CT3 + t] = (rv - mu) * inv * gg + bb;
}

// ------------- Kernel 5: fp32 WMMA GEMM (768x192) @ (192x1024) ---------------
// One wave per 16x16 output tile; 48 steps of v_wmma_f32_16x16x4_f32.
__global__ void k_gemm(const float* __restrict__ Lm, const float* __restrict__ Rm,
                       const float* __restrict__ WU, const float* __restrict__ bU,
                       const float* __restrict__ WV, const float* __restrict__ bV,
                       const float* __restrict__ sigma, float* __restrict__ out) {
    const int side = blockIdx.z;                  // 0 = left, 1 = right
    const float* A    = side ? Rm : Lm;
    const float* W    = side ? WV : WU;
    const float* bias = side ? bV : bU;
    float* O = out + (size_t)side * L * NOUT;

    const int wave = threadIdx.x >> 5;            // 0..3
    const int lane = threadIdx.x & 31;
    const int mt = blockIdx.y;                    // 0..47
    const int nt = blockIdx.x * 4 + wave;         // 0..63
    const int l16 = lane & 15;
    const int hi  = lane >> 4;                    // 0 or 1 (lane half)
    const int arow = mt * 16 + l16;               // A row this lane carries
    const int ncol = nt * 16 + l16;               // B/D column this lane carries

    // hoisted tile-invariant epilogue operands (no branches in writeback)
    const float bi = bias[ncol];
    const float sc = (side == 0) ? sigma[ncol >> 7] : 1.0f;

    v8f acc = {0.f, 0.f, 0.f, 0.f, 0.f, 0.f, 0.f, 0.f};
    for (int k0 = 0; k0 < CT3; k0 += 4) {
        int ka = k0 + 2 * hi;
        // A 16x4: lanes 0-15 hold K=k0,k0+1 ; lanes 16-31 hold K=k0+2,k0+3
        v2f a = *(const v2f*)(A + (size_t)arow * CT3 + ka);
        // B 4x16: V0 = row k0 (lanes 0-15) / row k0+2 (lanes 16-31); V1 = k0+1 / k0+3
        v2f bb;
        bb.x = W[(size_t)(ka + 0) * NOUT + ncol];
        bb.y = W[(size_t)(ka + 1) * NOUT + ncol];
        acc = __builtin_amdgcn_wmma_f32_16x16x4_f32(
            /*neg_a=*/false, a, /*neg_b=*/false, bb,
            /*c_mod=*/(short)0, acc, /*reuse_a=*/false, /*reuse_b=*/false);
    }
    // D 16x16: VGPR r -> row r (lanes 0-15) / row r+8 (lanes 16-31), col = lane&15
#pragma unroll
    for (int r = 0; r < 8; ++r) {
        int row = mt * 16 + r + 8 * hi;
        O[(size_t)row * NOUT + ncol] = (acc[r] + bi) * sc;
    }
}

extern "C" void kernel_launch(void* const* d_in, const int* in_sizes, int n_in,
                              void* d_out, int out_size, void* d_ws, size_t ws_size,
                              hipStream_t stream) {
    const float* feat  = (const float*)d_in[0];   // (1,768,768,64)
    const float* Wq    = (const float*)d_in[1];   // (64,1)
    const float* bq    = (const float*)d_in[2];   // (1,)
    const float* ln_g  = (const float*)d_in[3];   // (192,)
    const float* ln_b  = (const float*)d_in[4];   // (192,)
    const float* WU    = (const float*)d_in[5];   // (192,1024)
    const float* bU    = (const float*)d_in[6];   // (1024,)
    const float* WV    = (const float*)d_in[7];   // (192,1024)
    const float* bV    = (const float*)d_in[8];   // (1024,)
    const float* sigma = (const float*)d_in[9];   // (8,)
    float* out = (float*)d_out;

    // workspace layout (floats), ~3.95 MB total
    float* ws     = (float*)d_ws;
    float* logits = ws;                       // 768*768
    float* rmax   = logits + (size_t)L * L;   // 768
    float* rsum   = rmax + L;                 // 768
    float* cmax   = rsum + L;                 // 768
    float* csum   = cmax + L;                 // 768
    float* rowf   = csum + L;                 // 768*64
    float* colf   = rowf + (size_t)L * C;     // 768*64
    float* Lm     = colf + (size_t)L * C;     // 768*192
    float* Rm     = Lm + (size_t)L * CT3;     // 768*192

    // 1) logits
    k_logits<<<dim3((L * L) / 256), dim3(256), 0, stream>>>(feat, Wq, bq, logits);
    // 2) softmax stats for rows (axis=2) and cols (axis=1)
    k_stats<<<dim3(L, 2), dim3(256), 0, stream>>>(logits, rmax, rsum, cmax, csum);
    // 3) weighted sums (row_feat, col_feat)
    k_wsum<<<dim3(L), dim3(256), 0, stream>>>(feat, logits, rmax, rsum, rowf, 0);
    k_wsum<<<dim3(L), dim3(256), 0, stream>>>(feat, logits, cmax, csum, colf, 1);
    // 4) layernorm / build left & right
    k_ln<<<dim3(L), dim3(192), 0, stream>>>(feat, rowf, colf, ln_g, ln_b, Lm, Rm);
    // 5) WMMA GEMMs: (768x192)@(192x1024) for left (sigma-scaled) and right
    k_gemm<<<dim3(16, 48, 2), dim3(128), 0, stream>>>(Lm, Rm, WU, bU, WV, bV, sigma, out);
}